// YoloV3Decoder_64845416235378
// MI455X (gfx1250) — compile-verified
//
#include <hip/hip_runtime.h>

// ---------------------------------------------------------------------------
// YoloV3 decoder for MI455X (gfx1250).
// Memory-bound: ~49 MB in, ~7 MB out, trivial math -> optimize data movement.
// CDNA5 path: GLOBAL_LOAD_ASYNC_TO_LDS_B128 (ASYNCcnt) staging of each
// 128-anchor tile into LDS, issued as a straight run of 21 (+1 tail) async
// ops using the shared INST_OFFSET immediate (applied to both global and LDS
// addresses), then s_wait_asynccnt 0 + barrier. Per-lane decode reads LDS at
// word-stride 85 -> conflict-free on 64 banks (gcd(85,64)==1).
// wave32 / WGP sizing: 128 threads = 4 waves, 43.5 KB LDS per block.
// cpol=0 (RT): input (49MB) fits the 192MB L2 and is re-read across graph
// replays, so we WANT it cached; NT hints would hurt here.
// ---------------------------------------------------------------------------

namespace {
constexpr int   kNumAnchors      = 145152;     // NUM_ANCHORS
constexpr int   kNumCh           = 85;         // NUM_CH
constexpr int   kNumCls          = 80;         // classes (ch 5..84)
constexpr int   kAnchorsPerBlock = 128;
constexpr int   kThreads         = 128;        // 4 wave32
constexpr float kImgSize         = 1536.0f;

constexpr int kF4PerTile = kAnchorsPerBlock * kNumCh / 4;  // 2720 float4
constexpr int kFullPasses = kF4PerTile / kThreads;         // 21
constexpr int kTailLanes  = kF4PerTile % kThreads;         // 32
}

// Builtin expects int4-vector pointers: global (AS1) src, LDS (AS3) dst.
typedef int v4i __attribute__((ext_vector_type(4)));
typedef __attribute__((address_space(1))) v4i* g_v4i_ptr;
typedef __attribute__((address_space(3))) v4i* l_v4i_ptr;

// Probe the gfx1250 async-to-LDS builtins; host pass / older toolchains fall
// back to plain vector loads so compilation never breaks.
#if defined(__has_builtin)
#  if __has_builtin(__builtin_amdgcn_global_load_async_to_lds_b128)
#    define YV3_HAVE_ASYNC_LDS 1
#  endif
#  if __has_builtin(__builtin_amdgcn_s_wait_asynccnt)
#    define YV3_HAVE_WAIT_ASYNCCNT 1
#  endif
#endif

#if defined(__AMDGCN__)
#  if defined(YV3_HAVE_ASYNC_LDS)
#    pragma message("yolo_decode: device pass USING global_load_async_to_lds_b128")
#  else
#    pragma message("yolo_decode: device pass FALLBACK (no async-to-LDS builtin)")
#  endif
#endif

#if defined(YV3_HAVE_ASYNC_LDS)
// Compile-time-unrolled async copy: one address pair, immediate offsets.
// INST_OFFSET is added to BOTH the global and the LDS address (ISA 15.18.3),
// so consecutive passes just bump the literal by kThreads*16 = 2048 bytes.
template <int K>
struct AsyncTileCopy {
    static __device__ __forceinline__ void run(const float* g, float* l) {
        __builtin_amdgcn_global_load_async_to_lds_b128(
            (g_v4i_ptr)g, (l_v4i_ptr)l,
            /*offset=*/K * kThreads * 16, /*cpol=*/0);
        AsyncTileCopy<K + 1>::run(g, l);
    }
};
template <>
struct AsyncTileCopy<kFullPasses> {
    static __device__ __forceinline__ void run(const float*, float*) {}
};
#endif

__global__ __launch_bounds__(kThreads) void yolo_decode_kernel(
    const float* __restrict__ pred,   // [N, 85]
    float* __restrict__ out)          // [4N | N | N(int32) | 6N]
{
    __shared__ float tile[kAnchorsPerBlock * kNumCh];   // 43,520 bytes

    const int tid      = threadIdx.x;
    const int tileBase = blockIdx.x * kAnchorsPerBlock;
    const float* __restrict__ gsrc = pred + (size_t)tileBase * kNumCh;

    // ---- Stage tile into LDS, coalesced 16B per lane ----------------------
#if defined(YV3_HAVE_ASYNC_LDS)
    {
        const float* g = gsrc + 4 * tid;     // 16B aligned
        float*       l = tile + 4 * tid;     // 16B aligned
        AsyncTileCopy<0>::run(g, l);         // 21 async ops, literal offsets
        if (tid < kTailLanes) {              // last 32 float4s
            __builtin_amdgcn_global_load_async_to_lds_b128(
                (g_v4i_ptr)g, (l_v4i_ptr)l,
                /*offset=*/kFullPasses * kThreads * 16, /*cpol=*/0);
        }
    }
    // Each wave drains its own ASYNCcnt, then the workgroup barrier makes all
    // waves' LDS writes visible to everyone.
#  if defined(YV3_HAVE_WAIT_ASYNCCNT)
    __builtin_amdgcn_s_wait_asynccnt(0);
#  else
    asm volatile("s_wait_asynccnt 0" ::: "memory");
#  endif
    __syncthreads();
#else
    {
        const float4* __restrict__ g4 = reinterpret_cast<const float4*>(gsrc);
        float4* l4 = reinterpret_cast<float4*>(tile);
        for (int j = tid; j < kF4PerTile; j += kThreads) {
            l4[j] = g4[j];
        }
    }
    __syncthreads();
#endif

    // ---- Decode one anchor per thread out of LDS --------------------------
    const float* __restrict__ a = &tile[tid * kNumCh];

    // bboxes = predictions[..., :4] / IMG_SIZE   (divide first, like the ref)
    const float b0 = a[0] / kImgSize;
    const float b1 = a[1] / kImgSize;
    const float b2 = a[2] / kImgSize;
    const float b3 = a[3] / kImgSize;
    const float obj = a[4];

    // max / argmax over 80 classes (strict '>' => first-max, jnp.argmax rule)
    float bestConf = a[5];
    int   bestIdx  = 0;
#pragma unroll 8
    for (int c = 1; c < kNumCls; ++c) {
        const float v = a[5 + c];
        if (v > bestConf) { bestConf = v; bestIdx = c; }
    }

    const float x1 = b0 - b2 * 0.5f;
    const float y1 = b1 - b3 * 0.5f;
    const float x2 = b0 + b2 * 0.5f;
    const float y2 = b1 + b3 * 0.5f;
    const float score = obj * bestConf;

    const int aidx = tileBase + tid;

    // Output 0: bboxes [N,4]  (16B-aligned float4 store)
    reinterpret_cast<float4*>(out)[aidx] = make_float4(x1, y1, x2, y2);

    // Output 1: scores [N]
    out[(size_t)4 * kNumAnchors + aidx] = score;

    // Output 2: class_pred [N] -- int32 payload, store raw bits
    reinterpret_cast<int*>(out)[(size_t)5 * kNumAnchors + aidx] = bestIdx;

    // Output 3: detections [N,6] = {x1,y1,x2,y2,conf,(float)pred}
    // base offset 6N + 6*aidx is even -> 8B aligned, use float2 stores.
    float2* det2 = reinterpret_cast<float2*>(out + (size_t)6 * kNumAnchors
                                                 + (size_t)aidx * 6);
    det2[0] = make_float2(x1, y1);
    det2[1] = make_float2(x2, y2);
    det2[2] = make_float2(bestConf, (float)bestIdx);
}

extern "C" void kernel_launch(void* const* d_in, const int* in_sizes, int n_in,
                              void* d_out, int out_size, void* d_ws, size_t ws_size,
                              hipStream_t stream) {
    (void)in_sizes; (void)n_in; (void)d_ws; (void)ws_size; (void)out_size;

    const float* pred = (const float*)d_in[0];   // (1, 145152, 85) float32
    // d_in[1] = score_threshold: unused by the reference computation.
    float* out = (float*)d_out;                  // 12*N floats, all written

    static_assert(kNumAnchors % kAnchorsPerBlock == 0, "no tail blocks");
    const int grid = kNumAnchors / kAnchorsPerBlock;   // 1134 blocks

    yolo_decode_kernel<<<grid, kThreads, 0, stream>>>(pred, out);
}